// BasicConvolutionBlock_7868380086733
// MI455X (gfx1250) — compile-verified
//
#include <hip/hip_runtime.h>
#include <stdint.h>

#define CIN   32
#define COUT  64
#define KOFF  27
#define BN_EPS 1e-5f

#define WAVES_PER_BLOCK 8
#define ROWS_PER_WAVE   32
#define ROWS_PER_BLOCK  (WAVES_PER_BLOCK * ROWS_PER_WAVE)   // 256 rows per block
#define W_FLOATS        (KOFF * CIN * COUT)                 // 55296 floats = 221184 B (LDS-staged)
#define W_BYTES         (W_FLOATS * 4)
#define STAT_FLOATS     2048                                // 16 slots x 64 ch x {sum,sq}

typedef __attribute__((ext_vector_type(2))) float v2f;
typedef __attribute__((ext_vector_type(8))) float v8f;

// ---------------------------------------------------------------------------
// Kernel 1: async-staged weights + software-pipelined gather + fp32 WMMA GEMM
// ---------------------------------------------------------------------------
__global__ __launch_bounds__(256) void conv_wmma_kernel(
    const float* __restrict__ feats,    // [N, 32]
    const float* __restrict__ W,        // [27, 32, 64]
    const int64_t* __restrict__ nbr,    // [27, N], value N == sentinel
    const float* __restrict__ zrow,     // 32 zero floats (workspace)
    float* __restrict__ out,            // [N, 64] conv output
    float* __restrict__ partials,       // [gridDim.x, 128] {sum[64], sq[64]}
    int n)
{
    extern __shared__ float smem[];
    float* sW    = smem;                // W staged: [k][cin][cout]
    float* sStat = smem + W_FLOATS;     // [2][16 slots][64 ch]

    const int tid = threadIdx.x;

    // ---- CDNA5 async copy of all weights into LDS (221 KB), no VGPR round trip
    {
        // low 32 bits of the generic pointer == wave-relative LDS byte address
        const unsigned ldsBase = (unsigned)(uintptr_t)sW + (unsigned)tid * 16u;
        const unsigned gBase   = (unsigned)tid * 16u;
        #pragma unroll 2
        for (int i = 0; i < W_BYTES / (256 * 16); ++i) {      // 54 x 16B per thread
            unsigned lds = ldsBase + (unsigned)i * 4096u;
            unsigned g   = gBase   + (unsigned)i * 4096u;
            asm volatile("global_load_async_to_lds_b128 %0, %1, %2"
                         :: "v"(lds), "v"(g), "s"(W) : "memory");
        }
        asm volatile("s_wait_asynccnt 0x0" ::: "memory");
    }
    __syncthreads();

    const int  wave = tid >> 5;
    const int  lane = tid & 31;
    const int  lo   = lane & 15;        // row/col within 16
    const int  hi   = lane >> 4;        // half-wave selector
    const long base = (long)blockIdx.x * ROWS_PER_BLOCK + (long)wave * ROWS_PER_WAVE;

    // accumulators: two 16-row M-tiles x four 16-col N-tiles, fp32 C/D
    v8f accA[4], accB[4];
    const v8f vzero = {0.f, 0.f, 0.f, 0.f, 0.f, 0.f, 0.f, 0.f};
    #pragma unroll
    for (int t = 0; t < 4; ++t) { accA[t] = vzero; accB[t] = vzero; }

    const long r0  = base + lo;         // rows of M-tile 0
    const long r1  = base + 16 + lo;    // rows of M-tile 1
    const long nm1 = (long)n - 1;
    const long r0c = r0 < nm1 ? r0 : nm1;   // clamp for safe nbr loads on tail
    const long r1c = r1 < nm1 ? r1 : nm1;

    // A-fragment double buffer: 16x4 fp32 layout, lane lo holds cols {4c+2hi,+1}
    v2f a0[2][8], a1[2][8];

    auto gatherA = [&](int buf, long j0, long j1) {
        const bool v0 = (r0 < (long)n) && (j0 < (long)n);   // sentinel j==n -> zeros
        const bool v1 = (r1 < (long)n) && (j1 < (long)n);
        const float* p0 = (v0 ? (feats + j0 * CIN) : zrow) + 2 * hi;
        const float* p1 = (v1 ? (feats + j1 * CIN) : zrow) + 2 * hi;
        #pragma unroll
        for (int c = 0; c < 8; ++c) {
            a0[buf][c] = *(const v2f*)(p0 + 4 * c);
            a1[buf][c] = *(const v2f*)(p1 + 4 * c);
        }
    };

    // ---- software pipeline prologue: idx 2 deep, A-fragments 1 deep
    long j0 = nbr[r0c];
    long j1 = nbr[r1c];
    gatherA(0, j0, j1);                          // A[k=0]
    long jn0 = nbr[(long)n + r0c];               // idx for k=1 (KOFF >= 2)
    long jn1 = nbr[(long)n + r1c];

    for (int k = 0; k < KOFF; ++k) {
        const int cur = k & 1;
        const int nxt = cur ^ 1;

        // issue next gathers / index prefetch BEFORE this k's WMMA block
        // (uniform branches: EXEC stays all-1s through the WMMA region)
        if (k + 1 < KOFF) {
            gatherA(nxt, jn0, jn1);              // A[k+1] overlaps WMMA of k
            if (k + 2 < KOFF) {
                jn0 = nbr[(long)(k + 2) * n + r0c];
                jn1 = nbr[(long)(k + 2) * n + r1c];
            }
        }

        // B fragments from LDS: 4x16 fp32, vgpr0 = row 4c+2hi, vgpr1 = row +1
        const float* wk = sW + k * (CIN * COUT) + (2 * hi) * COUT + lo;
        #pragma unroll
        for (int c = 0; c < 8; ++c) {
            const float* wc = wk + (4 * c) * COUT;
            #pragma unroll
            for (int t = 0; t < 4; ++t) {
                v2f b;
                b.x = wc[16 * t];
                b.y = wc[COUT + 16 * t];
                accA[t] = __builtin_amdgcn_wmma_f32_16x16x4_f32(
                    false, a0[cur][c], false, b, (short)0, accA[t], false, false);
                accB[t] = __builtin_amdgcn_wmma_f32_16x16x4_f32(
                    false, a1[cur][c], false, b, (short)0, accB[t], false, false);
            }
        }
    }

    // store conv_out: C/D layout vgpr v -> row v + 8*hi, col = 16t + lo
    #pragma unroll
    for (int t = 0; t < 4; ++t) {
        #pragma unroll
        for (int v = 0; v < 8; ++v) {
            const long rowA = base + v + 8 * hi;
            const long rowB = base + 16 + v + 8 * hi;
            if (rowA < (long)n) out[rowA * COUT + 16 * t + lo] = accA[t][v];
            if (rowB < (long)n) out[rowB * COUT + 16 * t + lo] = accB[t][v];
        }
    }

    // deterministic per-block channel stats (no float atomics)
    const int slot = wave * 2 + hi;     // 16 slots
    #pragma unroll
    for (int t = 0; t < 4; ++t) {
        float s = 0.f, q = 0.f;
        #pragma unroll
        for (int v = 0; v < 8; ++v) {
            const long rowA = base + v + 8 * hi;
            const long rowB = base + 16 + v + 8 * hi;
            const float xA = (rowA < (long)n) ? accA[t][v] : 0.f;
            const float xB = (rowB < (long)n) ? accB[t][v] : 0.f;
            s += xA + xB;
            q += xA * xA + xB * xB;
        }
        sStat[slot * 64 + 16 * t + lo]        = s;
        sStat[1024 + slot * 64 + 16 * t + lo] = q;
    }
    __syncthreads();
    if (tid < 128) {
        const int ch    = tid & 63;
        const int which = tid >> 6;     // 0 = sum, 1 = sumsq
        float acc = 0.f;
        #pragma unroll
        for (int s2 = 0; s2 < 16; ++s2) acc += sStat[which * 1024 + s2 * 64 + ch];
        partials[(long)blockIdx.x * 128 + which * 64 + ch] = acc;
    }
}

// ---------------------------------------------------------------------------
// Kernel 2: reduce per-block partials (fixed order -> deterministic), BN coeffs
// ---------------------------------------------------------------------------
__global__ __launch_bounds__(64) void bn_stats_kernel(
    const float* __restrict__ partials, int nBlocks,
    const float* __restrict__ gamma, const float* __restrict__ beta,
    float* __restrict__ scale, float* __restrict__ shift, int n)
{
    const int c = threadIdx.x;          // 64 channels
    float s = 0.f, q = 0.f;
    for (int b = 0; b < nBlocks; ++b) {
        s += partials[(long)b * 128 + c];
        q += partials[(long)b * 128 + 64 + c];
    }
    const float invn = 1.f / (float)n;
    const float mean = s * invn;
    const float var  = fmaxf(q * invn - mean * mean, 0.f);   // biased var (matches jnp.var)
    const float inv  = rsqrtf(var + BN_EPS);
    const float sc   = gamma[c] * inv;
    scale[c] = sc;
    shift[c] = beta[c] - mean * sc;
}

// ---------------------------------------------------------------------------
// Kernel 3: in-place BN apply + ReLU, float4-vectorized.
// stride (grid*block) is a multiple of 16 float4s -> channel group fixed/thread
// ---------------------------------------------------------------------------
__global__ __launch_bounds__(256) void bn_relu_kernel(
    float* __restrict__ out,
    const float* __restrict__ scale, const float* __restrict__ shift,
    long total4)
{
    const long start  = (long)blockIdx.x * blockDim.x + threadIdx.x;
    const long stride = (long)gridDim.x * blockDim.x;
    const int  cb     = (int)((start * 4) & 63);
    const float4 sc = *(const float4*)(scale + cb);
    const float4 sh = *(const float4*)(shift + cb);
    for (long e = start; e < total4; e += stride) {
        float4 x = ((const float4*)out)[e];
        float4 y;
        y.x = fmaxf(0.f, x.x * sc.x + sh.x);
        y.y = fmaxf(0.f, x.y * sc.y + sh.y);
        y.z = fmaxf(0.f, x.z * sc.z + sh.z);
        y.w = fmaxf(0.f, x.w * sc.w + sh.w);
        ((float4*)out)[e] = y;
    }
}

// ---------------------------------------------------------------------------
extern "C" void kernel_launch(void* const* d_in, const int* in_sizes, int n_in,
                              void* d_out, int out_size, void* d_ws, size_t ws_size,
                              hipStream_t stream)
{
    const float*   feats = (const float*)d_in[0];     // [N,32] f32
    const float*   W     = (const float*)d_in[1];     // [27,32,64] f32
    const float*   gamma = (const float*)d_in[2];     // [64] f32
    const float*   beta  = (const float*)d_in[3];     // [64] f32
    const int64_t* nbr   = (const int64_t*)d_in[4];   // [27,N] i64
    float*         out   = (float*)d_out;             // [N,64] f32

    const int n = in_sizes[0] / CIN;

    // workspace layout (floats): [0..31] zero row | [32..95] scale | [96..159] shift
    //                            [160..) per-block partials [nBlocks][128]
    float* wsf      = (float*)d_ws;
    float* zrow     = wsf;
    float* scale    = wsf + 32;
    float* shift    = wsf + 96;
    float* partials = wsf + 160;

    hipMemsetAsync(zrow, 0, 32 * sizeof(float), stream);   // graph-capturable

    const int nBlocks = (n + ROWS_PER_BLOCK - 1) / ROWS_PER_BLOCK;
    const size_t smemBytes = (size_t)(W_FLOATS + STAT_FLOATS) * sizeof(float);  // ~229 KB < 320 KB

    conv_wmma_kernel<<<nBlocks, 256, smemBytes, stream>>>(feats, W, nbr, zrow, out, partials, n);
    bn_stats_kernel<<<1, 64, 0, stream>>>(partials, nBlocks, gamma, beta, scale, shift, n);

    const long total4 = ((long)n * COUT) / 4;
    bn_relu_kernel<<<2048, 256, 0, stream>>>(out, scale, shift, total4);
}